// HardTripletLoss_884763263444
// MI455X (gfx1250) — compile-verified
//
#include <hip/hip_runtime.h>
#include <hip/hip_bf16.h>
#include <math.h>

// HardTripletLoss for MI455X (gfx1250), compile-only tuning.
//
// Phase 1 (dominant, ~614 MB streamed, memory-bound, ~26us roofline @23.3TB/s):
//   relations[r] = sqrt( sum_d (emb[r,d] - attr[r,d] + 1e-6)^2 )
//   Per 16-row tile: Tensor Data Mover DMAs raw emb/attr 16x100 chunks into
//   wave-private LDS buffers (double-buffered, TENSORcnt-tracked), the wave
//   computes squared diffs per A-fragment and row-sums on the matrix core via
//   V_WMMA_F32_16X16X4_F32 with B = ones (75 K-steps of 4 per tile).
// Phase 2: per-column (over B=256) masked max/min reductions (1 MB, L2-hot).
// Phase 3: scalar finalize.

typedef float v2f __attribute__((ext_vector_type(2)));
typedef float v8f __attribute__((ext_vector_type(8)));
typedef unsigned int u32x4 __attribute__((ext_vector_type(4)));
typedef unsigned int u32x8 __attribute__((ext_vector_type(8)));

#define D_DIM 300
#define CHUNK 100                 // K-chunk width (floats); 3 chunks per row
#define NCHUNK 3
#define ROWS_PER_TILE 16
#define WAVES_PER_BLOCK 2
#define CHUNK_FLOATS (ROWS_PER_TILE * CHUNK)   // 1600 floats = 6400 B

// Low 32 bits of the flat shared-aperture address are the LDS byte offset.
__device__ __forceinline__ unsigned lds_off(const void* p)
{
    return (unsigned)(uintptr_t)p;
}

// Issue one TDM load of a 16xCHUNK f32 tile (row stride D_DIM) into LDS.
// 2D tensor -> descriptor groups 0 and 1 only (VADDR2/VADDR3 = NULL form).
__device__ __forceinline__ void tdm_load_tile(const float* gptr,
                                              unsigned lds_byte_off,
                                              unsigned rows_total)
{
    const unsigned long long ga = (unsigned long long)(uintptr_t)gptr;

    u32x4 g0;
    g0[0] = 0x1u;                                   // count=1, user descriptor
    g0[1] = lds_byte_off;                           // lds_addr (bytes)
    g0[2] = (unsigned)(ga & 0xFFFFFFFFu);           // global_addr[31:0]
    g0[3] = ((unsigned)(ga >> 32) & 0x01FFFFFFu)    // global_addr[56:32]
            | (2u << 30);                           // type = 2 ("image")

    u32x8 g1;
    g1[0] = (2u << 16);                             // data_size = 4 bytes
    g1[1] = ((unsigned)D_DIM & 0xFFFFu) << 16;      // tensor_dim0[15:0]
    g1[2] = ((rows_total & 0xFFFFu) << 16)          // tensor_dim1[15:0]
            | (((unsigned)D_DIM >> 16) & 0xFFFFu);  // tensor_dim0[31:16]
    g1[3] = ((rows_total >> 16) & 0xFFFFu)          // tensor_dim1[31:16]
            | ((unsigned)CHUNK << 16);              // tile_dim0 = CHUNK
    g1[4] = (unsigned)ROWS_PER_TILE;                // tile_dim1 = 16, tile_dim2 = 0
    g1[5] = (unsigned)D_DIM;                        // tensor_dim0_stride[31:0]
    g1[6] = 0;                                      // stride hi / dim1_stride lo
    g1[7] = 0;

    asm volatile("tensor_load_to_lds %0, %1"
                 :
                 : "s"(g0), "s"(g1)
                 : "memory");
}

__global__ __launch_bounds__(64) void triplet_relations_kernel(
    const float* __restrict__ attributes,
    const float* __restrict__ embeddings,
    float* __restrict__ relations,
    int rows)
{
    // [wave][tensor: 0=emb 1=attr][double-buffer][16*100 floats] = 51.2 KB
    __shared__ float stage[WAVES_PER_BLOCK][2][2][CHUNK_FLOATS];

    const int lane = threadIdx.x & 31;
    const int wave = threadIdx.x >> 5;
    const int tile = blockIdx.x * WAVES_PER_BLOCK + wave;   // exact grid
    const int r0   = tile * ROWS_PER_TILE;
    const size_t gbase = (size_t)r0 * D_DIM;

    // Prime the pipeline: chunk 0 into buffer 0 (TENSORcnt += 2).
    tdm_load_tile(embeddings + gbase, lds_off(&stage[wave][0][0][0]), (unsigned)rows);
    tdm_load_tile(attributes + gbase, lds_off(&stage[wave][1][0][0]), (unsigned)rows);

    // WMMA A layout (32-bit, 16x4): lanes 0-15 -> M=lane, VGPR{0,1}=K{0,1};
    //                               lanes 16-31 -> M=lane-16, VGPR{0,1}=K{2,3}.
    const int half  = lane >> 4;
    const int m     = lane & 15;
    const int lbase = m * CHUNK + half * 2;   // stride 100 -> conflict-free banks

    v8f acc = {};
    v2f ones = {1.0f, 1.0f};

    for (int c = 0; c < NCHUNK; ++c) {
        if (c + 1 < NCHUNK) {
            const size_t goff = gbase + (size_t)(c + 1) * CHUNK;
            tdm_load_tile(embeddings + goff,
                          lds_off(&stage[wave][0][(c + 1) & 1][0]), (unsigned)rows);
            tdm_load_tile(attributes + goff,
                          lds_off(&stage[wave][1][(c + 1) & 1][0]), (unsigned)rows);
            __builtin_amdgcn_s_wait_tensorcnt(2);   // chunk c's 2 loads done
        } else {
            __builtin_amdgcn_s_wait_tensorcnt(0);   // drain
        }
        asm volatile("" ::: "memory");              // order DS reads after TDM wait

        const float* ebuf = &stage[wave][0][c & 1][lbase];
        const float* abuf = &stage[wave][1][c & 1][lbase];
        #pragma unroll 5
        for (int k = 0; k < CHUNK / 4; ++k) {
            v2f e = *(const v2f*)(ebuf + k * 4);
            v2f a = *(const v2f*)(abuf + k * 4);
            v2f d = e - a + 1e-6f;                  // torch pairwise_distance eps
            v2f s = d * d;
            acc = __builtin_amdgcn_wmma_f32_16x16x4_f32(
                /*neg_a=*/false, s,
                /*neg_b=*/false, ones,
                /*c_mod=*/(short)0, acc,
                /*reuse_a=*/false, /*reuse_b=*/false);
        }
    }

    // D layout: VGPR v holds M=v (lanes 0-15) / M=v+8 (lanes 16-31).
    // Column N=0 lives in lanes 0 and 16.
    if (m == 0) {
        const int rbase = r0 + half * 8;
        #pragma unroll
        for (int v = 0; v < 8; ++v) {
            relations[rbase + v] = sqrtf(acc[v]);
        }
    }
}

__global__ void triplet_cols_kernel(const float* __restrict__ relations,
                                    const int* __restrict__ labels,
                                    float* __restrict__ accum,
                                    int B, int C)
{
    __shared__ int lab[256];
    const int t = threadIdx.x;
    if (t < B) lab[t] = labels[t];
    __syncthreads();

    const int c = blockIdx.x * blockDim.x + t;
    float triplet = 0.0f, hard = 0.0f;
    if (c < C) {
        // Pass 1: hardest positive (masked max; relations >= 0 so 0-init matches
        // the reference's max over relations*mask) and global max per column.
        float maxAll = 0.0f;
        float maxPos = 0.0f;
        for (int b = 0; b < B; ++b) {
            float r = relations[b * C + c];
            maxAll = fmaxf(maxAll, r);
            maxPos = (lab[b] == c) ? fmaxf(maxPos, r) : maxPos;
        }
        // Pass 2: hardest negative: min over b of (r + maxAll * mask).
        float minNeg = 3.402823466e38f;
        for (int b = 0; b < B; ++b) {
            float r = relations[b * C + c];
            r += (lab[b] == c) ? maxAll : 0.0f;
            minNeg = fminf(minNeg, r);
        }
        triplet = fmaxf(maxPos - minNeg + 1.0f, 0.0f);
        hard    = (triplet > 1e-16f) ? 1.0f : 0.0f;
    }
    atomicAdd(&accum[0], triplet);
    atomicAdd(&accum[1], hard);
}

__global__ void triplet_init_kernel(float* __restrict__ accum)
{
    accum[0] = 0.0f;
    accum[1] = 0.0f;
}

__global__ void triplet_finalize_kernel(const float* __restrict__ accum,
                                        float* __restrict__ out)
{
    out[0] = accum[0] / (accum[1] + 1e-16f);
}

extern "C" void kernel_launch(void* const* d_in, const int* in_sizes, int n_in,
                              void* d_out, int out_size, void* d_ws, size_t ws_size,
                              hipStream_t stream)
{
    const float* attributes = (const float*)d_in[0];
    const float* embeddings = (const float*)d_in[1];
    const int*   labels     = (const int*)d_in[2];

    const int B = in_sizes[2];                 // 256
    const int rows = in_sizes[0] / D_DIM;      // 256000
    const int C = rows / B;                    // 1000

    float* relations = (float*)d_ws;           // rows floats
    float* accum     = relations + rows;       // [sum, count]

    triplet_init_kernel<<<1, 1, 0, stream>>>(accum);

    const int tiles  = rows / ROWS_PER_TILE;             // 16000
    const int blocks = tiles / WAVES_PER_BLOCK;          // 8000, exact
    triplet_relations_kernel<<<blocks, WAVES_PER_BLOCK * 32, 0, stream>>>(
        attributes, embeddings, relations, rows);

    const int cb = (C + 255) / 256;
    triplet_cols_kernel<<<cb, 256, 0, stream>>>(relations, labels, accum, B, C);

    triplet_finalize_kernel<<<1, 1, 0, stream>>>(accum, (float*)d_out);
}